// TemplateMatching_5935644803374
// MI455X (gfx1250) — compile-verified
//
#include <hip/hip_runtime.h>
#include <cstdint>
#include <cstddef>

typedef __attribute__((ext_vector_type(4))) float f4;

#ifndef __has_builtin
#define __has_builtin(x) 0
#endif

// --- CDNA5 async global->LDS path (ASYNCcnt) --------------------------------
#if __has_builtin(__builtin_amdgcn_s_wait_asynccnt)
#define WAIT_ASYNCCNT(n) __builtin_amdgcn_s_wait_asynccnt(n)
#else
#define WAIT_ASYNCCNT(n) asm volatile("s_wait_asynccnt %0" ::"i"(n) : "memory")
#endif

// GV mode: vdst = per-lane LDS byte address, vaddr = 64-bit global address.
// IOFFSET is added to BOTH the LDS and global addresses (ISA §15.18 async
// pseudocode), so one base address pair serves all 9 chunk issues.
// TH_LOAD_NT: plane data is read exactly once -> don't pollute L2.
__device__ __forceinline__ void async_load_b128_to_lds(uint32_t lds_byte_addr,
                                                       const void* gptr,
                                                       int imm_off) {
  asm volatile("global_load_async_to_lds_b128 %0, %1, off offset:%2 th:TH_LOAD_NT"
               :: "v"(lds_byte_addr), "v"((unsigned long long)gptr), "i"(imm_off)
               : "memory");
}
// ----------------------------------------------------------------------------

constexpr int kC = 256;
constexpr int kH = 96;
constexpr int kW = 96;
constexpr int kHW = kH * kW;                 // 9216 floats = 36 KB per plane
constexpr int kThreads = 256;                // 8 wave32 waves
constexpr int kVec = kHW / 4 / kThreads;     // 9 float4 chunks per thread
constexpr int kPPB = 4;                      // planes per block (pipelined)

__global__ __launch_bounds__(kThreads)
void tm_fused_kernel(const float* __restrict__ feat,
                     const float* __restrict__ ex,
                     const float* __restrict__ scale,
                     float* __restrict__ out) {
  extern __shared__ float smem[];            // 2*kHW plane buffers + 32 reduce slots
  float* red = smem + 2 * kHW;

  const int tid = threadIdx.x;
  const int plane0 = blockIdx.x * kPPB;      // kPPB divides kC -> one batch per block
  const int b = plane0 / kC;

  // Per-thread base LDS byte address (buffer 0); buffer 1 is +kHW*4 bytes.
  const uint32_t lds0 = (uint32_t)(uintptr_t)(void*)smem + (uint32_t)tid * 16u;

  // Kick off plane 0 -> buf0 immediately; hide latency behind box/mask setup.
  {
    const char* g = (const char*)(feat + (size_t)plane0 * kHW) + (size_t)tid * 16;
#pragma unroll
    for (int k = 0; k < kVec; ++k)
      async_load_b128_to_lds(lds0, g, k * 4096);
  }

  // ---- Box geometry (uniform per block) ----
  const float cx1 = fminf(fmaxf(ex[b * 4 + 0], 0.f), 1.f);
  const float cy1 = fminf(fmaxf(ex[b * 4 + 1], 0.f), 1.f);
  const float cx2 = fminf(fmaxf(ex[b * 4 + 2], 0.f), 1.f);
  const float cy2 = fminf(fmaxf(ex[b * 4 + 3], 0.f), 1.f);
  const float x1 = floorf(cx1 * (float)kW), y1 = floorf(cy1 * (float)kH);
  const float x2 = ceilf(cx2 * (float)kW),  y2 = ceilf(cy2 * (float)kH);
  const float inv_count = 1.0f / fmaxf((x2 - x1) * (y2 - y1), 1.0f);
  const float sc = scale[0];

  // ---- Per-thread spatial mask for its 9 float4 chunks (same every plane) ----
  f4 m[kVec];
#pragma unroll
  for (int k = 0; k < kVec; ++k) {
    const int j = k * kThreads + tid;              // float4 index within plane
    const float hf = (float)(j / (kW / 4));
    const float wb = (float)((j % (kW / 4)) * 4);
    const float ym = (hf >= y1 && hf < y2) ? 1.f : 0.f;
#pragma unroll
    for (int e = 0; e < 4; ++e) {
      const float wf = wb + (float)e;
      m[k][e] = (wf >= x1 && wf < x2) ? ym : 0.f;
    }
  }

  // ---- Pipelined plane loop: DMA next plane while processing current ----
  for (int i = 0; i < kPPB; ++i) {
    if (i + 1 < kPPB) {
      const char* g =
          (const char*)(feat + (size_t)(plane0 + i + 1) * kHW) + (size_t)tid * 16;
      const uint32_t ldsn = lds0 + (uint32_t)(((i + 1) & 1) * kHW * 4);
#pragma unroll
      for (int k = 0; k < kVec; ++k)
        async_load_b128_to_lds(ldsn, g, k * 4096);
      WAIT_ASYNCCNT(kVec);   // first 9 (current plane) retired; next 9 in flight
    } else {
      WAIT_ASYNCCNT(0);
    }
    __syncthreads();          // plane i fully resident in LDS for all 8 waves

    // Pull this thread's chunk into registers once; use it twice.
    const float* bp = smem + (i & 1) * kHW + tid * 4;
    f4 v[kVec];
    float partial = 0.f;
#pragma unroll
    for (int k = 0; k < kVec; ++k) {
      v[k] = *(const f4*)(bp + k * 1024);
      partial = fmaf(v[k].x, m[k].x, partial);
      partial = fmaf(v[k].y, m[k].y, partial);
      partial = fmaf(v[k].z, m[k].z, partial);
      partial = fmaf(v[k].w, m[k].w, partial);
    }

    // wave32 butterfly reduce, then 8 wave partials through LDS
#pragma unroll
    for (int off = 16; off > 0; off >>= 1)
      partial += __shfl_xor(partial, off, 32);
    if ((tid & 31) == 0) red[tid >> 5] = partial;
    __syncthreads();
    float tot = 0.f;
#pragma unroll
    for (int w = 0; w < kThreads / 32; ++w) tot += red[w];
    const float factor = tot * inv_count * sc;   // /(1+1e-14) == *1.0f in fp32

    // Streamed output: non-temporal B128 stores (don't pollute L2)
    float* o = out + (size_t)(plane0 + i) * kHW + tid * 4;
#pragma unroll
    for (int k = 0; k < kVec; ++k) {
      f4 r = v[k] * factor;
      __builtin_nontemporal_store(r, (f4*)(o + k * 1024));
    }
    __syncthreads();          // all reads of red[] and buf[i&1] done before reuse
  }
}

extern "C" void kernel_launch(void* const* d_in, const int* in_sizes, int n_in,
                              void* d_out, int out_size, void* d_ws, size_t ws_size,
                              hipStream_t stream) {
  const float* feat  = (const float*)d_in[0];   // (bs, 256, 96, 96) f32
  const float* ex    = (const float*)d_in[1];   // (bs, 1, 4) f32
  const float* scale = (const float*)d_in[2];   // (1,) f32
  float* out = (float*)d_out;

  const int bs = in_sizes[1] / 4;               // 32
  const int planes = bs * kC;                   // 8192
  const int grid = planes / kPPB;               // 2048 blocks
  const size_t shmem = (size_t)(2 * kHW + 32) * sizeof(float);  // ~74 KB

  tm_fused_kernel<<<grid, kThreads, shmem, stream>>>(feat, ex, scale, out);
}